// Beamformer_33225867002149
// MI455X (gfx1250) — compile-verified
//
#include <hip/hip_runtime.h>

typedef __attribute__((ext_vector_type(2))) float v2f;
typedef __attribute__((ext_vector_type(8))) float v8f;

#define N_TX   3
#define N_AX   2048
#define N_EL   128
#define N_Z    512
#define N_X    256
#define N_PIX  (N_Z * N_X)

// delays = dist / SOUND_SPEED * FS
#define DELAY_SCALE (20000000.0f / 1540.0f)

// ---------------------------------------------------------------------------
// One-shot layout transform: (tx, ax, el) -> (tx, el, ax).
// After this, a wave of 16 neighboring pixels gathering nearby delays for a
// fixed element reads contiguous addresses (stride 4B in ax) -> 1-2 cachelines
// per gather instead of 16 lines. Data is 3 MB, L2-resident.
// ---------------------------------------------------------------------------
__global__ __launch_bounds__(256) void transpose_kernel(
    const float* __restrict__ in, float* __restrict__ out) {
  int tid = blockIdx.x * blockDim.x + threadIdx.x;   // 0 .. 786431
  // input linear index = ((tx*N_AX)+ax)*N_EL + el  (coalesced read)
  int el   = tid & (N_EL - 1);
  int rest = tid >> 7;               // tx*N_AX + ax
  int ax   = rest & (N_AX - 1);
  int tx   = rest >> 11;
  out[(tx * N_EL + el) * N_AX + ax] = in[tid];
}

// ---------------------------------------------------------------------------
// DAS beamformer. One wave32 per 16-pixel tile (16 consecutive x positions in
// one z-row). Per 4-element chunk each lane computes 2 masked interpolated
// samples (pixel = lane&15, elements chunk*4 + (lane>>4)*2 + {0,1}) which is
// exactly the V_WMMA_F32_16X16X4_F32 A-operand layout. B = ones, so the WMMA
// accumulates the element/tx sum per pixel in the f32 matrix pipe, freeing the
// (bottlenecked) VALU of the accumulate FMA. EXEC is all-1s at every WMMA.
// ---------------------------------------------------------------------------
template <bool TRANS>
__global__ __launch_bounds__(256) void beamform_kernel(
    const float* __restrict__ data,     // TRANS: (tx, el, ax) else (tx, ax, el)
    const float* __restrict__ grid,     // (N_PIX, 3)
    const float* __restrict__ probe,    // (N_EL, 3)
    const float* __restrict__ angles,   // (N_TX)
    float* __restrict__ out)            // (N_PIX)
{
  __shared__ float s_ex[N_EL];
  int t = threadIdx.x;
  if (t < N_EL) s_ex[t] = probe[t * 3 + 0];
  __syncthreads();

  int lane  = t & 31;
  int wave  = t >> 5;
  int group = blockIdx.x * 8 + wave;    // 16-pixel tile index
  int m     = lane & 15;                // pixel row within tile (A-matrix M)
  int half  = lane >> 4;                // 0 -> K=0,1 ; 1 -> K=2,3
  int pix   = group * 16 + m;

  float xp = grid[pix * 3 + 0];
  float zp = grid[pix * 3 + 2];
  float z2 = zp * zp;
  float halfAper = 0.5f * zp;           // aperture/2 with F_NUMBER = 1

  // Transmit-path distances (element-invariant per pixel; 3 sin + 3 cos total).
  float dtx[N_TX];
#pragma unroll
  for (int tx = 0; tx < N_TX; ++tx) {
    float a = angles[tx];
    dtx[tx] = xp * sinf(a) + zp * cosf(a);
  }

  v8f c = {};                           // per-pixel running sum (16x16 D/C)
  const v2f bones = {1.0f, 1.0f};       // B = all ones -> row-sum reduction

  for (int chunk = 0; chunk < N_EL / 4; ++chunk) {
    int el0 = chunk * 4 + half * 2;

    // tx-invariant receive distance + apodization mask (hoisted: 2 sqrt/chunk)
    float rx[2], msk[2];
#pragma unroll
    for (int j = 0; j < 2; ++j) {
      float ex = s_ex[el0 + j];
      float dx = xp - ex;
      rx[j]  = sqrtf(dx * dx + z2);
      msk[j] = (fabsf(dx) <= halfAper) ? 1.0f : 0.0f;   // branchless cndmask
    }

#pragma unroll
    for (int tx = 0; tx < N_TX; ++tx) {
      v2f a;
#pragma unroll
      for (int j = 0; j < 2; ++j) {
        float w  = (dtx[tx] + rx[j]) * DELAY_SCALE;     // fractional delay
        int   d0 = (int)floorf(w);
        d0 = min(max(d0, 0), N_AX - 1);
        int   d1 = min(d0 + 1, N_AX - 1);
        int   el = el0 + j;
        float s0, s1;
        if (TRANS) {
          int base = (tx * N_EL + el) * N_AX;
          s0 = data[base + d0];
          s1 = data[base + d1];
        } else {
          s0 = data[(tx * N_AX + d0) * N_EL + el];
          s1 = data[(tx * N_AX + d1) * N_EL + el];
        }
        float d0f = (float)d0, d1f = (float)d1;
        float smp = (d1f - w) * s0 + (w - d0f) * s1;    // linear interp
        a[j] = smp * msk[j];
      }
      // D = A(16x4 samples) x B(ones) + C : accumulates sum over 4 elements
      // for all 16 pixels; f32 precision throughout.
      c = __builtin_amdgcn_wmma_f32_16x16x4_f32(
          /*neg_a=*/false, a, /*neg_b=*/false, bones,
          /*c_mod=*/(short)0, c, /*reuse_a=*/false, /*reuse_b=*/false);
    }
  }

  // D layout: VGPR r, lanes 0-15 -> M=r ; lanes 16-31 -> M=8+r (all columns
  // equal). Lane 0 stores pixels 0..7, lane 16 stores pixels 8..15.
  if (m == 0) {
    int base = group * 16 + half * 8;
#pragma unroll
    for (int r = 0; r < 8; ++r) out[base + r] = c[r];
  }
}

extern "C" void kernel_launch(void* const* d_in, const int* in_sizes, int n_in,
                              void* d_out, int out_size, void* d_ws, size_t ws_size,
                              hipStream_t stream) {
  const float* data   = (const float*)d_in[0];   // (1,3,2048,128,1) f32
  const float* grid   = (const float*)d_in[1];   // (131072,3)       f32
  const float* probe  = (const float*)d_in[2];   // (128,3)          f32
  const float* angles = (const float*)d_in[3];   // (3,)             f32
  float* out = (float*)d_out;                    // (1,512,256)      f32

  const int n_groups = N_PIX / 16;               // 8192 wave-tiles
  const int n_blocks = n_groups / 8;             // 8 waves (256 thr) per block

  size_t need = (size_t)N_TX * N_AX * N_EL * sizeof(float);  // 3 MB
  if (ws_size >= need) {
    float* dataT = (float*)d_ws;
    transpose_kernel<<<(N_TX * N_AX * N_EL) / 256, 256, 0, stream>>>(data, dataT);
    beamform_kernel<true><<<n_blocks, 256, 0, stream>>>(dataT, grid, probe,
                                                        angles, out);
  } else {
    beamform_kernel<false><<<n_blocks, 256, 0, stream>>>(data, grid, probe,
                                                         angles, out);
  }
}